// GCNEncoder_13580686590282
// MI455X (gfx1250) — compile-verified
//
#include <hip/hip_runtime.h>

typedef __attribute__((ext_vector_type(2))) float v2f;
typedef __attribute__((ext_vector_type(8))) float v8f;

// ---------------- degree / normalization ----------------

__global__ void fill_f32(float* __restrict__ p, int n, float v) {
  int i = blockIdx.x * blockDim.x + threadIdx.x;
  if (i < n) p[i] = v;
}

__global__ void deg_edges(const int* __restrict__ dst, float* __restrict__ deg, int e) {
  int i = blockIdx.x * blockDim.x + threadIdx.x;
  if (i < e) atomicAdd(&deg[dst[i]], 1.0f);
}

__global__ void deg_to_dinv(float* __restrict__ d, int n) {
  int i = blockIdx.x * blockDim.x + threadIdx.x;
  if (i < n) {
    float v = d[i];
    d[i] = (v > 0.0f) ? rsqrtf(v) : 0.0f;
  }
}

// ---------------- fp32 WMMA GEMM: Out[n,OUTC] = A[n,K] @ W[K,OUTC] -------------
// Grid: x = row-tile groups (8 row tiles of 16 per block), y = column tile (32 cols).
// Weight panel is staged in LDS in K-pairwise-packed layout:
//     wlds[(k>>1)*96 + col*2 + (k&1)]
// so each lane's B fragment {W[k0][n], W[k0+1][n]} is one contiguous 8-byte
// ds_load_b64 directly into the even-aligned VGPR pair WMMA consumes (no repack
// moves). Stride 96 (== 32 mod 64 banks): in one WMMA step lanes 0-15 read kpair
// p (even -> dword banks 0-31) and lanes 16-31 read kpair p+1 (+96 -> banks
// 32-63), so the wave's 64 dwords hit 64 distinct banks -- conflict-free.
// Per K-step: one global_load_b64 (A fragment) + two ds_load_b64 (B fragments)
// + two independent V_WMMA_F32_16X16X4_F32 accumulators sharing A.
// A layout (ISA 7.12.2): lanes 0-15 -> K=0,1 ; lanes 16-31 -> K=2,3 of row lane&15.
// C/D layout: VGPR v -> row v (lanes 0-15) / v+8 (lanes 16-31), col = lane&15.

template <int K, int OUTC>
__global__ void gemm_wmma_f32(const float* __restrict__ A, const float* __restrict__ W,
                              float* __restrict__ Out, int nRows) {
  constexpr int KPSTRIDE = 96;                   // dword stride per k-pair row
  __shared__ float wlds[(K / 2) * KPSTRIDE];

  const int lane = threadIdx.x & 31;
  const int waveInBlk = threadIdx.x >> 5;
  const int ct = blockIdx.y;                     // column tile (32 cols)

  // Cooperative stage of the K x 32 weight panel into packed LDS layout.
  for (int idx = threadIdx.x; idx < K * 32; idx += blockDim.x) {
    const int k  = idx >> 5;
    const int cc = idx & 31;
    wlds[(k >> 1) * KPSTRIDE + cc * 2 + (k & 1)] =
        W[(size_t)k * OUTC + ct * 32 + cc];
  }
  __syncthreads();

  const int rowTiles = nRows >> 4;
  const int rt = blockIdx.x * (blockDim.x >> 5) + waveInBlk;
  if (rt >= rowTiles) return;                    // uniform per wave -> EXEC stays full

  const int mn = lane & 15;                      // row (A) / col (B,C,D)
  const int kh = lane >> 4;                      // K-half select
  const float* aRow = A + (size_t)(rt * 16 + mn) * K;
  const float* bBase = wlds + kh * KPSTRIDE + mn * 2;  // lane-local B base

  v8f c0 = {};
  v8f c1 = {};

  #pragma unroll 8
  for (int kb = 0; kb < K; kb += 4) {
    const int k0 = kb + 2 * kh;
    v2f a  = *(const v2f*)(aRow + k0);                       // 8B-aligned
    v2f b0 = *(const v2f*)(bBase + (kb >> 1) * KPSTRIDE);      // cols 0-15 half
    v2f b1 = *(const v2f*)(bBase + (kb >> 1) * KPSTRIDE + 32); // cols 16-31 half
    c0 = __builtin_amdgcn_wmma_f32_16x16x4_f32(false, a, false, b0, (short)0, c0, false, false);
    c1 = __builtin_amdgcn_wmma_f32_16x16x4_f32(false, a, false, b1, (short)0, c1, false, false);
  }

  float* oBase = Out + (size_t)(rt * 16 + 8 * kh) * OUTC + (size_t)ct * 32 + mn;
  #pragma unroll
  for (int v = 0; v < 8; ++v) {
    oBase[(size_t)v * OUTC]      = c0[v];
    oBase[(size_t)v * OUTC + 16] = c1[v];
  }
}

// ---------------- self-loop init: Out[i,:] = H[i,:] * dinv[i]^2 ----------------
// (doubles as the zero-initialization of the aggregation buffer)

__global__ void self_init(const float* __restrict__ H, const float* __restrict__ dinv,
                          float* __restrict__ Out, int total, int D) {
  int i = blockIdx.x * blockDim.x + threadIdx.x;
  if (i >= total) return;
  float di = dinv[i / D];
  Out[i] = H[i] * di * di;
}

// ---------------- wave-per-edge scatter: Out[dst] += H[src]*dinv[src]*dinv[dst] --
// Gather stays L2-resident (feature matrix ~102MB < 192MB L2); float4 loads,
// per-dword global_atomic_add_f32 resolved in L2 atomic units.

template <int D>
__global__ void scatter_edges(const float* __restrict__ H, const int* __restrict__ srcIdx,
                              const int* __restrict__ dstIdx, const float* __restrict__ dinv,
                              float* __restrict__ Out, int nEdges) {
  const int lane = threadIdx.x & 31;
  const int e = blockIdx.x * (blockDim.x >> 5) + (threadIdx.x >> 5);
  if (e >= nEdges) return;                       // uniform per wave
  const int s = srcIdx[e];
  const int d = dstIdx[e];
  const float nrm = dinv[s] * dinv[d];
  constexpr int CPL = D / 32;                    // channels per lane: 8 (D=256) / 4 (D=128)
  const float* hp = H + (size_t)s * D + lane * CPL;
  float* op = Out + (size_t)d * D + lane * CPL;
  #pragma unroll
  for (int i = 0; i < CPL; i += 4) {
    float4 v = *(const float4*)(hp + i);
    atomicAdd(op + i + 0, v.x * nrm);
    atomicAdd(op + i + 1, v.y * nrm);
    atomicAdd(op + i + 2, v.z * nrm);
    atomicAdd(op + i + 3, v.w * nrm);
  }
}

// ---------------- epilogues ----------------

__global__ void bias_relu(float* __restrict__ X, const float* __restrict__ b,
                          int total, int D) {
  int i = blockIdx.x * blockDim.x + threadIdx.x;
  if (i >= total) return;
  X[i] = fmaxf(X[i] + b[i % D], 0.0f);
}

__global__ void final_relu(float* __restrict__ Out, const float* __restrict__ b,
                           const float* __restrict__ x0, int total, int D) {
  int i = blockIdx.x * blockDim.x + threadIdx.x;
  if (i >= total) return;
  Out[i] = fmaxf(Out[i] + b[i % D] + x0[i], 0.0f);
}

// ---------------- launch ----------------

extern "C" void kernel_launch(void* const* d_in, const int* in_sizes, int n_in,
                              void* d_out, int out_size, void* d_ws, size_t ws_size,
                              hipStream_t stream) {
  const float* x  = (const float*)d_in[0];
  const float* W0 = (const float*)d_in[1];
  const float* b0 = (const float*)d_in[2];
  const float* W1 = (const float*)d_in[3];
  const float* b1 = (const float*)d_in[4];
  const float* W2 = (const float*)d_in[5];
  const float* b2 = (const float*)d_in[6];
  const int*   ei = (const int*)d_in[7];

  const int IN_C = 128, C1 = 256, C2 = 128;
  const int N = in_sizes[0] / IN_C;
  const int E = in_sizes[7] / 2;
  const int* srcIdx = ei;        // edge_index[0]
  const int* dstIdx = ei + E;    // edge_index[1]

  // workspace layout (floats): dinv[N] | bufA[N*256] | x0[N*256] | bufD[N*128]
  float* ws   = (float*)d_ws;
  float* dinv = ws;
  float* bufA = dinv + N;                  // H0 / H1 / H2 staging
  float* x0   = bufA + (size_t)N * C1;     // layer-0 activations (residual)
  float* bufD = x0   + (size_t)N * C1;     // layer-1 activations
  float* outp = (float*)d_out;

  const int T = 256;
  auto cdiv = [](long a, long b) { return (int)((a + b - 1) / b); };
  const int rowTiles = N / 16;
  const int rtBlocks = cdiv(rowTiles, T / 32);

  // symmetric normalization D^{-1/2}(A+I)D^{-1/2}
  fill_f32<<<cdiv(N, T), T, 0, stream>>>(dinv, N, 1.0f);             // self-loop degree
  deg_edges<<<cdiv(E, T), T, 0, stream>>>(dstIdx, dinv, E);
  deg_to_dinv<<<cdiv(N, T), T, 0, stream>>>(dinv, N);

  // ---- layer 0: 128 -> 256, ReLU ----
  {
    dim3 g(rtBlocks, C1 / 32);
    gemm_wmma_f32<128, 256><<<g, T, 0, stream>>>(x, W0, bufA, N);
    self_init<<<cdiv((long)N * C1, T), T, 0, stream>>>(bufA, dinv, x0, N * C1, C1);
    scatter_edges<256><<<cdiv(E, T / 32), T, 0, stream>>>(bufA, srcIdx, dstIdx, dinv, x0, E);
    bias_relu<<<cdiv((long)N * C1, T), T, 0, stream>>>(x0, b0, N * C1, C1);
  }
  // ---- layer 1: 256 -> 128, ReLU ----
  {
    dim3 g(rtBlocks, C2 / 32);
    gemm_wmma_f32<256, 128><<<g, T, 0, stream>>>(x0, W1, bufA, N);
    self_init<<<cdiv((long)N * C2, T), T, 0, stream>>>(bufA, dinv, bufD, N * C2, C2);
    scatter_edges<128><<<cdiv(E, T / 32), T, 0, stream>>>(bufA, srcIdx, dstIdx, dinv, bufD, E);
    bias_relu<<<cdiv((long)N * C2, T), T, 0, stream>>>(bufD, b1, N * C2, C2);
  }
  // ---- layer 2: 128 -> 256, residual + ReLU ----
  {
    dim3 g(rtBlocks, C1 / 32);
    gemm_wmma_f32<128, 256><<<g, T, 0, stream>>>(bufD, W2, bufA, N);
    self_init<<<cdiv((long)N * C1, T), T, 0, stream>>>(bufA, dinv, outp, N * C1, C1);
    scatter_edges<256><<<cdiv(E, T / 32), T, 0, stream>>>(bufA, srcIdx, dstIdx, dinv, outp, E);
    final_relu<<<cdiv((long)N * C1, T), T, 0, stream>>>(outp, b2, x0, N * C1, C1);
  }
}